// CTCLoss_39737037423186
// MI455X (gfx1250) — compile-verified
//
#include <hip/hip_runtime.h>
#include <hip/hip_bf16.h>
#include <stdint.h>

// Problem constants (match reference)
#define T_ 128
#define B_ 128
#define C_ 6625
#define S_ 32
#define L_ 65            // 2*S+1
#define NEGV (-1e30f)
#define GCOLS 33         // blank + 32 labels (distinct classes per sample)

// C_ = 6625 floats = 1656 x b128 chunks + 1 x b32 tail
#define CHUNKS_B128 1656u

// ---------------------------------------------------------------------------
// Kernel 1: per (t,b) row -> logsumexp over C, gather 33 needed classes.
// Streams the row into LDS with CDNA5 async global->LDS loads (ASYNCcnt),
// reduces from LDS, gathers label classes from LDS (no 2nd global pass).
// ---------------------------------------------------------------------------
__global__ __launch_bounds__(256) void k_lse_gather(const float* __restrict__ preds,
                                                    const int* __restrict__ labels,
                                                    float* __restrict__ lpg) {
    __shared__ __align__(16) float srow[6656];   // 6625 used, padded
    __shared__ float red[8];
    __shared__ float bcast[2];

    const int tid  = threadIdx.x;
    const int lane = tid & 31;
    const int wid  = tid >> 5;
    const int row  = blockIdx.x;          // row = t*B + b
    const int b    = row & (B_ - 1);
    const float* base = preds + (size_t)row * C_;   // uniform -> SGPR pair

    // ---- async stream row into LDS (per-lane b128 transfers) ----
    for (unsigned c = tid; c < CHUNKS_B128; c += 256u) {
        unsigned lds = (unsigned)(uintptr_t)(&srow[c * 4u]);
        unsigned off = c * 16u;
        asm volatile("global_load_async_to_lds_b128 %0, %1, %2"
                     :: "v"(lds), "v"(off), "s"(base) : "memory");
    }
    if (tid == 0) {  // tail element 6624
        unsigned lds = (unsigned)(uintptr_t)(&srow[6624]);
        unsigned off = 6624u * 4u;
        asm volatile("global_load_async_to_lds_b32 %0, %1, %2"
                     :: "v"(lds), "v"(off), "s"(base) : "memory");
    }
    asm volatile("s_wait_asynccnt 0x0" ::: "memory");
    __syncthreads();

    // ---- block max over C ----
    float m = -3.0e38f;
    for (int i = tid; i < C_; i += 256) m = fmaxf(m, srow[i]);
    for (int o = 16; o; o >>= 1) m = fmaxf(m, __shfl_xor(m, o, 32));
    if (lane == 0) red[wid] = m;
    __syncthreads();
    if (wid == 0) {
        float x = (lane < 8) ? red[lane] : -3.0e38f;
        for (int o = 4; o; o >>= 1) x = fmaxf(x, __shfl_xor(x, o, 32));
        if (lane == 0) bcast[0] = x;
    }
    __syncthreads();
    const float M = bcast[0];

    // ---- block sum of exp(x - M) ----
    float acc = 0.0f;
    for (int i = tid; i < C_; i += 256) acc += __expf(srow[i] - M);
    for (int o = 16; o; o >>= 1) acc += __shfl_xor(acc, o, 32);
    if (lane == 0) red[wid] = acc;
    __syncthreads();
    if (wid == 0) {
        float x = (lane < 8) ? red[lane] : 0.0f;
        for (int o = 4; o; o >>= 1) x += __shfl_xor(x, o, 32);
        if (lane == 0) bcast[1] = x;
    }
    __syncthreads();
    const float lse = M + __logf(bcast[1]);

    // ---- gather 33 classes from LDS: col 0 = blank, col 1+j = labels[b][j] ----
    if (tid < GCOLS) {
        int cls = (tid == 0) ? 0 : labels[b * S_ + (tid - 1)];
        lpg[(size_t)row * GCOLS + tid] = srow[cls] - lse;
    }
}

// ---------------------------------------------------------------------------
// Kernel 2: CTC forward DP per sample. One block of 96 threads per b,
// states s = 0..64 live in LDS; recurrence over t with split barriers.
// ---------------------------------------------------------------------------
__global__ __launch_bounds__(96) void k_ctc_dp(const float* __restrict__ lpg,
                                               const int* __restrict__ labels,
                                               const int* __restrict__ preds_size,
                                               const int* __restrict__ labels_len,
                                               float* __restrict__ losses) {
    __shared__ float alpha[L_];
    const int b = blockIdx.x;
    const int s = threadIdx.x;
    const bool act = (s < L_);
    const bool odd = (s & 1) != 0;
    const int  j   = (s - 1) >> 1;

    bool skip = false;                    // a2 allowed: odd state and label != label[j-1]
    if (act && odd) {
        int lab  = labels[b * S_ + j];
        int labm = (j > 0) ? labels[b * S_ + j - 1] : -1;
        skip = (lab != labm);
    }
    const int gcol = odd ? (1 + j) : 0;
    const int ll   = labels_len[b];
    const int Tb   = preds_size[b];       // block-uniform: "freeze" == stop early
    const float* lpb = lpg + (size_t)b * GCOLS;

    if (act) {
        float a = NEGV;
        if (s == 0)      a = lpb[0];
        else if (s == 1) a = (ll > 0) ? lpb[1] : NEGV;
        alpha[s] = a;
    }
    __syncthreads();

    for (int t = 1; t < Tb; ++t) {
        float a0 = NEGV, a1 = NEGV, a2 = NEGV;
        if (act) {
            a0 = alpha[s];
            if (s >= 1) a1 = alpha[s - 1];
            if (s >= 2 && skip) a2 = alpha[s - 2];
        }
        __syncthreads();
        if (act) {
            float m  = fmaxf(a0, fmaxf(a1, a2));
            float e  = __expf(a0 - m) + __expf(a1 - m) + __expf(a2 - m);
            float lp = lpb[(size_t)t * (B_ * GCOLS) + gcol];
            alpha[s] = m + __logf(e) + lp;
        }
        __syncthreads();
    }

    if (s == 0) {
        int il = 2 * ll;
        int ip = (il - 1 > 0) ? il - 1 : 0;
        float al = alpha[il], ap = alpha[ip];
        float m  = fmaxf(al, ap);
        losses[b] = -(m + __logf(__expf(al - m) + __expf(ap - m)));
    }
}

// ---------------------------------------------------------------------------
// Kernel 3: deterministic final reduction: out = sum(losses) / 128
// ---------------------------------------------------------------------------
__global__ __launch_bounds__(128) void k_reduce(const float* __restrict__ losses,
                                                float* __restrict__ out) {
    __shared__ float red[4];
    const int tid = threadIdx.x, lane = tid & 31, wid = tid >> 5;
    float v = losses[tid];
    for (int o = 16; o; o >>= 1) v += __shfl_xor(v, o, 32);
    if (lane == 0) red[wid] = v;
    __syncthreads();
    if (tid == 0) out[0] = (red[0] + red[1] + red[2] + red[3]) * (1.0f / 128.0f);
}

// ---------------------------------------------------------------------------
extern "C" void kernel_launch(void* const* d_in, const int* in_sizes, int n_in,
                              void* d_out, int out_size, void* d_ws, size_t ws_size,
                              hipStream_t stream) {
    const float* preds      = (const float*)d_in[0];   // [T,B,C] f32
    const int*   labels     = (const int*)d_in[1];     // [B,S]  i32
    const int*   preds_size = (const int*)d_in[2];     // [B]    i32
    const int*   labels_len = (const int*)d_in[3];     // [B]    i32

    float* lpg    = (float*)d_ws;                                  // [T,B,33] ~2.1 MB
    size_t lpgsz  = ((size_t)T_ * B_ * GCOLS * sizeof(float) + 255) & ~(size_t)255;
    float* losses = (float*)((char*)d_ws + lpgsz);                 // [B]

    k_lse_gather<<<T_ * B_, 256, 0, stream>>>(preds, labels, lpg);
    k_ctc_dp<<<B_, 96, 0, stream>>>(lpg, labels, preds_size, labels_len, losses);
    k_reduce<<<1, 128, 0, stream>>>(losses, (float*)d_out);
}